// CSAtt_71511205479164
// MI455X (gfx1250) — compile-verified
//
#include <hip/hip_runtime.h>
#include <hip/hip_bf16.h>
#include <math.h>

// Sizes fixed by the reference
#define BATCH 64
#define CH    512
#define HH    28
#define WW    28
#define PLANE (HH*WW)        // 784
#define POOL  49             // 7x7 pooled
#define KP    52             // K padded to multiple of 4 (WMMA f32 K=4 steps)
#define RDIM  32             // CH/16
#define NTHR  256            // 8 waves of 32

typedef float v2f __attribute__((ext_vector_type(2)));
typedef float v4f __attribute__((ext_vector_type(4)));
typedef float v8f __attribute__((ext_vector_type(8)));

__device__ __forceinline__ float block_sum(float v, float* s_red) {
    const int lane = threadIdx.x & 31;
    const int wv   = threadIdx.x >> 5;
    #pragma unroll
    for (int o = 16; o > 0; o >>= 1) v += __shfl_down(v, o, 32);
    if (lane == 0) s_red[wv] = v;
    __syncthreads();
    if (wv == 0) {
        float r = (lane < 8) ? s_red[lane] : 0.0f;
        #pragma unroll
        for (int o = 4; o > 0; o >>= 1) r += __shfl_down(r, o, 32);
        if (lane == 0) s_red[0] = r;
    }
    __syncthreads();
    float r = s_red[0];
    __syncthreads();           // allow s_red reuse by the next call
    return r;
}

__global__ __launch_bounds__(NTHR, 1)
void csatt_stats_kernel(const float* __restrict__ x,
                        const float* __restrict__ wD, const float* __restrict__ bD,
                        const float* __restrict__ wU, const float* __restrict__ bU,
                        float* __restrict__ d_scale) {
    __shared__ float s_xap[CH][KP];   // 106496 B  pooled features, zero-padded K
    __shared__ float s_sq[CH];        // per-channel sum of squares of xap row
    __shared__ float s_z[CH];         // gap, then z
    __shared__ float s_t[CH];         // accum t[i] = sum_j z_j sim[j,i]
    __shared__ float s_h[RDIM];       // MLP hidden
    __shared__ float s_red[16];

    const int b    = blockIdx.x;
    const int tid  = threadIdx.x;
    const int lane = tid & 31;
    const int wv   = tid >> 5;
    const int half = lane >> 4;       // 0: lanes 0-15, 1: lanes 16-31
    const int lr   = lane & 15;

    // ---------- Phase 1: 4x4 block-average pool + gap + sq ----------
    // wave wv owns channels [wv*64, wv*64+64)
    for (int ci = 0; ci < CH / 8; ++ci) {
        const int c = wv * (CH / 8) + ci;
        const float* xp = x + ((size_t)(b * CH + c)) * PLANE;
        float gsum = 0.0f, ssum = 0.0f;
        for (int pp = lane; pp < KP; pp += 32) {
            if (pp < POOL) {
                const int pr = pp / 7, pc = pp % 7;
                const float* bp = xp + pr * 4 * WW + pc * 4;
                v4f r0 = *(const v4f*)(bp);
                v4f r1 = *(const v4f*)(bp + WW);
                v4f r2 = *(const v4f*)(bp + 2 * WW);
                v4f r3 = *(const v4f*)(bp + 3 * WW);
                v4f rs = r0 + r1 + r2 + r3;
                float s16 = rs.x + rs.y + rs.z + rs.w;
                float pooled = s16 * (1.0f / 16.0f);
                s_xap[c][pp] = pooled;
                gsum += s16;
                ssum += pooled * pooled;
            } else {
                s_xap[c][pp] = 0.0f;   // zero K-padding (safe for gram & sq)
            }
        }
        #pragma unroll
        for (int o = 16; o > 0; o >>= 1) {
            gsum += __shfl_down(gsum, o, 32);
            ssum += __shfl_down(ssum, o, 32);
        }
        if (lane == 0) {
            s_z[c]  = gsum * (1.0f / (float)PLANE);   // gap
            s_sq[c] = ssum;
        }
    }
    __syncthreads();

    // ---------- Phase 2: z = (gap - mu) / std(ddof=1); init t ----------
    {
        float g0 = s_z[tid], g1 = s_z[tid + 256];
        float sum   = block_sum(g0 + g1, s_red);
        float sumsq = block_sum(g0 * g0 + g1 * g1, s_red);
        float mu  = sum * (1.0f / (float)CH);
        float var = (sumsq - (float)CH * mu * mu) * (1.0f / (float)(CH - 1));
        float sd  = sqrtf(fmaxf(var, 0.0f));
        s_z[tid]       = (g0 - mu) / sd;
        s_z[tid + 256] = (g1 - mu) / sd;
        s_t[tid] = 0.0f;
        s_t[tid + 256] = 0.0f;
    }
    __syncthreads();

    // ---------- Phase 3 (pass A): mean of d over the 512x512 gram ----------
    // 32x32 grid of 16x16 tiles; wave wv takes tiles wv, wv+8, ...
    float sumd = 0.0f;
    for (int tile = wv; tile < 1024; tile += 8) {
        const int i0 = (tile >> 5) * 16, j0 = (tile & 31) * 16;
        const int rowA = i0 + lr, rowB = j0 + lr;
        const int koff = 2 * half;
        v8f acc = {};
        #pragma unroll
        for (int k = 0; k < KP; k += 4) {
            v2f a  = *(const v2f*)&s_xap[rowA][k + koff];
            v2f bb = *(const v2f*)&s_xap[rowB][k + koff];
            acc = __builtin_amdgcn_wmma_f32_16x16x4_f32(
                false, a, false, bb, (short)0, acc, false, false);
        }
        #pragma unroll
        for (int rr = 0; rr < 8; ++rr) {
            const int mi = i0 + rr + 8 * half;
            const int nj = j0 + lr;
            float g  = acc[rr];
            float d2 = s_sq[mi] + s_sq[nj] - 2.0f * g;
            float d  = sqrtf(d2);
            d = (d != d) ? 0.0f : d;    // NaN (sqrt of negative) -> 0
            d = fmaxf(d, 0.0f);
            sumd += d;
        }
    }
    const float mean_d = block_sum(sumd, s_red) * (1.0f / ((float)CH * (float)CH));

    // ---------- Phase 4 (pass B): sim, S = sum(sim), t[i] = sum_j z_j sim[j,i] ----------
    float Sloc = 0.0f;
    for (int tile = wv; tile < 1024; tile += 8) {
        const int i0 = (tile >> 5) * 16, j0 = (tile & 31) * 16;
        const int rowA = i0 + lr, rowB = j0 + lr;
        const int koff = 2 * half;
        v8f acc = {};
        #pragma unroll
        for (int k = 0; k < KP; k += 4) {
            v2f a  = *(const v2f*)&s_xap[rowA][k + koff];
            v2f bb = *(const v2f*)&s_xap[rowB][k + koff];
            acc = __builtin_amdgcn_wmma_f32_16x16x4_f32(
                false, a, false, bb, (short)0, acc, false, false);
        }
        const int nj = j0 + lr;
        const float sqj = s_sq[nj];
        float tl = 0.0f;
        #pragma unroll
        for (int rr = 0; rr < 8; ++rr) {
            const int mi = i0 + rr + 8 * half;
            float g  = acc[rr];
            float d2 = s_sq[mi] + sqj - 2.0f * g;
            float d  = sqrtf(d2);
            d = (d != d) ? 0.0f : d;
            d = fmaxf(d, 0.0f);
            float l2  = expf(-d / (mean_d + 1e-10f));
            float wn  = sqrtf(s_sq[mi]) * sqrtf(sqj);
            float cs  = g / fmaxf(wn, 1e-10f);
            cs = (cs != cs) ? 0.0f : cs;
            cs = fmaxf(cs, 0.0f);
            float sim = cs * l2;
            if (mi == nj) sim = 0.0f;   // zero diagonal
            Sloc += sim;
            tl   += s_z[mi] * sim;
        }
        atomicAdd(&s_t[nj], tl);
    }
    const float S = block_sum(Sloc, s_red);   // syncs inside cover the LDS atomics

    // ---------- Phase 5: local_mi, normalize -> ch (in s_t) ----------
    {
        const float inv = 1.0f / (S + 1e-10f);
        float lm0 = s_z[tid]       * s_t[tid]       * inv;
        float lm1 = s_z[tid + 256] * s_t[tid + 256] * inv;
        float sum   = block_sum(lm0 + lm1, s_red);
        float sumsq = block_sum(lm0 * lm0 + lm1 * lm1, s_red);
        float m   = sum * (1.0f / (float)CH);
        float var = (sumsq - (float)CH * m * m) * (1.0f / (float)(CH - 1));
        float sd  = sqrtf(fmaxf(var, 0.0f));
        s_t[tid]       = (lm0 - m) / (sd + 1e-12f);
        s_t[tid + 256] = (lm1 - m) / (sd + 1e-12f);
    }
    __syncthreads();

    // ---------- Phase 6: h = relu(ch @ wD^T + bD) ----------
    if (tid < RDIM) s_h[tid] = bD[tid];
    __syncthreads();
    {
        const int r   = tid & 31;      // output index
        const int seg = tid >> 5;      // 8 segments of 64 channels
        float part = 0.0f;
        const float* wrow = wD + (size_t)r * CH;
        #pragma unroll 4
        for (int c = seg * 64; c < seg * 64 + 64; ++c) part += s_t[c] * wrow[c];
        atomicAdd(&s_h[r], part);
    }
    __syncthreads();
    if (tid < RDIM) s_h[tid] = fmaxf(s_h[tid], 0.0f);
    __syncthreads();

    // ---------- Phase 7: att = h @ wU^T + bU; scale = sigmoid(att) ----------
    for (int c = tid; c < CH; c += NTHR) {
        float att = bU[c];
        const float* wrow = wU + (size_t)c * RDIM;
        #pragma unroll
        for (int r = 0; r < RDIM; ++r) att += s_h[r] * wrow[r];
        d_scale[b * CH + c] = 1.0f / (1.0f + expf(-att));
    }
}

// Streaming apply: out = x * scale[b,c], float4 vectorized (B128 per lane).
// x loads are regular-temporal (want L2 hits from the stats kernel's pass);
// out stores are non-temporal (write-once data must not evict x from L2).
__global__ __launch_bounds__(NTHR, 4)
void csatt_apply_kernel(const float* __restrict__ x,
                        const float* __restrict__ d_scale,
                        float* __restrict__ out) {
    const int idx = blockIdx.x * NTHR + threadIdx.x;      // float4 index
    const int total4 = (BATCH * CH * PLANE) / 4;          // 6,422,528
    if (idx >= total4) return;
    const int plane = idx / (PLANE / 4);                  // (b*CH + c)
    const float s = d_scale[plane];
    v4f v = *((const v4f*)x + idx);
    v = v * s;
    __builtin_nontemporal_store(v, (v4f*)out + idx);
}

extern "C" void kernel_launch(void* const* d_in, const int* in_sizes, int n_in,
                              void* d_out, int out_size, void* d_ws, size_t ws_size,
                              hipStream_t stream) {
    const float* x  = (const float*)d_in[0];
    const float* wD = (const float*)d_in[1];
    const float* bD = (const float*)d_in[2];
    const float* wU = (const float*)d_in[3];
    const float* bU = (const float*)d_in[4];
    float* out     = (float*)d_out;
    float* d_scale = (float*)d_ws;     // BATCH*CH floats = 128 KB

    csatt_stats_kernel<<<BATCH, NTHR, 0, stream>>>(x, wD, bD, wU, bU, d_scale);

    const int total4 = (BATCH * CH * PLANE) / 4;
    const int nblk = (total4 + NTHR - 1) / NTHR;
    csatt_apply_kernel<<<nblk, NTHR, 0, stream>>>(x, d_scale, out);
}